// EvolvedNet_80942953661021
// MI455X (gfx1250) — compile-verified
//
#include <hip/hip_runtime.h>
#include <cstdint>

#define NUM_NODES   64
#define NUM_INPUTS  16
#define NUM_OUTPUTS 8
#define NUM_EDGES   256
#define BATCH       8192

#define BLOCK     64                      // lanes (batch columns) per block; 2 wave32 waves
#define CHUNK     8                       // edges per software-pipelined chunk
#define DUMMY     NUM_NODES               // scratch node row for padding edges
#define ROWS      (NUM_NODES + 1)         // 64 real nodes + 1 dummy row
#define MAX_SLOTS (NUM_EDGES * CHUNK)     // worst-case padded schedule length
#define SCHED_OFF 16                      // sched[0]=n_slots, data at sched+16 (64B aligned)

// gfx1250 native tanh (tanh-insts).
#if defined(__has_builtin)
#  if __has_builtin(__builtin_amdgcn_tanhf)
#    define FAST_TANH(v) __builtin_amdgcn_tanhf(v)
#  endif
#endif
#ifndef FAST_TANH
#  define FAST_TANH(v) tanhf(v)
#endif

// ---------------------------------------------------------------------------
// One-time dependency-levelization of the edge list (single thread, O(E)).
//
// level[e] = max( lastWrite[src[e]] + 1,   // RAW: must read in a later chunk
//                 lastWrite[dst[e]],       // WAW: same chunk ok (adds issue in
//                                          //      original index order)
//                 lastRead [dst[e]] )      // WAR: same chunk ok (all loads
//                                          //      issue before all adds)
//
// Stable counting sort by level + pad each level to a multiple of CHUNK with
// no-op edges (w=0 -> tanh(0)=0 added to the dummy row). Every conflicting
// edge pair keeps its sequential order, and the wave's LDS pipe is in-order,
// so the chunked execution is bitwise identical to the sequential reference.
// ---------------------------------------------------------------------------
__global__ void evonet_schedule_kernel(const int*   __restrict__ src,
                                       const int*   __restrict__ dst,
                                       const float* __restrict__ w,
                                       int*         __restrict__ sched)
{
    __shared__ int lvl[NUM_EDGES];
    __shared__ int lw[NUM_NODES];
    __shared__ int lr[NUM_NODES];
    __shared__ int start[NUM_EDGES + 1];   // padded level start offsets
    __shared__ int cur[NUM_EDGES];         // per-level emit cursor

    if (threadIdx.x != 0) return;
    float* pw = (float*)(sched + SCHED_OFF + MAX_SLOTS);

    for (int n = 0; n < NUM_NODES; ++n) { lw[n] = -1; lr[n] = -1; }
    for (int e = 0; e < NUM_EDGES; ++e) cur[e] = 0;   // reuse as level counts

    int maxlvl = 0;
    for (int e = 0; e < NUM_EDGES; ++e) {
        const int s = src[e], d = dst[e];
        int L = lw[s] + 1;
        if (lw[d] > L) L = lw[d];
        if (lr[d] > L) L = lr[d];
        lvl[e] = L;
        if (L > maxlvl) maxlvl = L;
        if (lr[s] < L) lr[s] = L;
        if (lw[d] < L) lw[d] = L;
        cur[L]++;
    }

    // Padded level starts.
    start[0] = 0;
    for (int L = 0; L <= maxlvl; ++L) {
        const int padded = (cur[L] + (CHUNK - 1)) & ~(CHUNK - 1);
        start[L + 1] = start[L] + padded;
    }
    const int n_slots = start[maxlvl + 1];

    // Fill everything with dummy no-op edges, then scatter real edges
    // stably (index order within a level preserves same-dst add order).
    for (int i = 0; i < n_slots; ++i) {
        sched[SCHED_OFF + i] = 0 | (DUMMY << 8);
        pw[i] = 0.0f;
    }
    for (int L = 0; L <= maxlvl; ++L) cur[L] = start[L];
    for (int e = 0; e < NUM_EDGES; ++e) {
        const int p = cur[lvl[e]]++;
        sched[SCHED_OFF + p] = src[e] | (dst[e] << 8);
        pw[p] = w[e];
    }
    sched[0] = n_slots;
}

// ---------------------------------------------------------------------------
// Main relaxation kernel. Per-lane-private column state in LDS
// (vals[row][lane], uniform row index -> conflict-free DS banking).
// Chunks of 8 conflict-free edges: 8 pipelined ds_load_b32, one wait,
// 8x (v_mul + v_tanh_f32), 8x ds_add_f32 (no-return LDS float atomic).
// ---------------------------------------------------------------------------
__global__ __launch_bounds__(BLOCK)
void evonet_exec_kernel(const float* __restrict__ x,
                        const int*   __restrict__ sched,
                        float*       __restrict__ out)
{
    __shared__ float vals[ROWS * BLOCK];

    const int t   = threadIdx.x;
    const int col = blockIdx.x * BLOCK + t;

    // Stage the 16 input rows straight into LDS via gfx1250 async DMA.
    for (int i = 0; i < NUM_INPUTS; ++i) {
        uint32_t lds_off   = (uint32_t)(uintptr_t)&vals[i * BLOCK + t];
        const float* gaddr = x + i * BATCH + col;
        asm volatile("global_load_async_to_lds_b32 %0, %1, off"
                     :: "v"(lds_off), "v"(gaddr)
                     : "memory");
    }
    // Zero hidden/output/dummy rows.
    #pragma unroll
    for (int i = NUM_INPUTS; i < ROWS; ++i)
        vals[i * BLOCK + t] = 0.0f;
    asm volatile("s_wait_asynccnt 0x0" ::: "memory");

    const int    n_slots = sched[0];                                  // uniform
    const int*   psd     = sched + SCHED_OFF;                         // s_load
    const float* pw      = (const float*)(sched + SCHED_OFF + MAX_SLOTS);

    for (int pass = 0; pass < NUM_NODES; ++pass) {
        for (int c = 0; c < n_slots; c += CHUNK) {
            int   sd[CHUNK];   // uniform -> SGPRs
            float wc[CHUNK];   // uniform -> SGPRs
            float v[CHUNK];

            // Phase 1: independent loads (levelization guarantees no RAW
            // inside a chunk). Compiler cannot hoist these above the previous
            // chunk's atomics (may-alias) -> cross-level order preserved.
            #pragma unroll
            for (int i = 0; i < CHUNK; ++i) {
                sd[i] = psd[c + i];
                wc[i] = pw[c + i];
                v[i]  = vals[(sd[i] & 0xFF) * BLOCK + t];
            }
            // Phase 2: tanh + accumulate. Same-dst adds issue in original
            // edge order -> bitwise identical to the sequential reference.
            #pragma unroll
            for (int i = 0; i < CHUNK; ++i) {
                atomicAdd(&vals[(sd[i] >> 8) * BLOCK + t],
                          FAST_TANH(v[i] * wc[i]));
            }
        }
    }

    // Emit tanh of output nodes [16, 24).
    #pragma unroll
    for (int o = 0; o < NUM_OUTPUTS; ++o)
        out[o * BATCH + col] = FAST_TANH(vals[(NUM_INPUTS + o) * BLOCK + t]);
}

extern "C" void kernel_launch(void* const* d_in, const int* in_sizes, int n_in,
                              void* d_out, int out_size, void* d_ws, size_t ws_size,
                              hipStream_t stream) {
    const float* x   = (const float*)d_in[0];  // [16, 8192]
    const float* w   = (const float*)d_in[1];  // [256]
    const int*   src = (const int*)  d_in[2];  // [256]
    const int*   dst = (const int*)  d_in[3];  // [256]
    float*       out = (float*)      d_out;    // [8, 8192]
    int*         ws  = (int*)        d_ws;     // schedule: ~16.5 KB used

    // 1) Build the conflict-free chunked schedule (deterministic, tiny).
    hipLaunchKernelGGL(evonet_schedule_kernel, dim3(1), dim3(32), 0, stream,
                       src, dst, w, ws);

    // 2) Run the relaxation over all 8192 columns.
    hipLaunchKernelGGL(evonet_exec_kernel, dim3(BATCH / BLOCK), dim3(BLOCK),
                       0, stream, x, ws, out);
}